// VectorQuantiser_57896159150485
// MI455X (gfx1250) — compile-verified
//
#include <hip/hip_runtime.h>
#include <hip/hip_bf16.h>

// Problem constants (from reference): B=16, D=256, H=W=32, K=8192
//   N = B*H*W = 16384 rows, z layout (B, D, H, W): z[b*262144 + d*1024 + hw]
//   d_out = [z_q : 4194304 f32][indices : 16384 (as f32)][commit_loss : 1 f32]
//   d_ws  = [cbnorm : 8192 f32][idx_i : 16384 i32]

typedef __attribute__((ext_vector_type(2))) float v2f;
typedef __attribute__((ext_vector_type(8))) float v8f;

#define VQ_K 8192
#define VQ_D 256
#define VQ_N 16384
#define TILE_CODES 16       // codes per LDS tile (one WMMA N-tile)
#define N_TILES (VQ_K / TILE_CODES)          // 512
#define LDS_STRIDE 260      // 256 dims + pad: 4c/4c+2 banks -> conflict-free b64 reads
#define ROWS_PER_BLOCK 128  // 8 waves * 16 rows

// ---------------------------------------------------------------------------
// 1) codebook squared norms: one wave per code
// ---------------------------------------------------------------------------
__global__ __launch_bounds__(256) void cb_norms_kernel(const float* __restrict__ cb,
                                                       float* __restrict__ cbnorm) {
  const int wave = threadIdx.x >> 5;
  const int lane = threadIdx.x & 31;
  const int code = blockIdx.x * 8 + wave;
  const float* row = cb + (size_t)code * VQ_D;
  float s = 0.0f;
#pragma unroll
  for (int d = 0; d < VQ_D / 32; ++d) {
    float x = row[d * 32 + lane];
    s += x * x;
  }
#pragma unroll
  for (int off = 16; off > 0; off >>= 1) s += __shfl_xor(s, off, 32);
  if (lane == 0) cbnorm[code] = s;
}

// ---------------------------------------------------------------------------
// 2) fused GEMM (z . e^T via v_wmma_f32_16x16x4_f32) + running argmin
//    grid = N / ROWS_PER_BLOCK = 128 blocks, 256 threads (8 waves)
// ---------------------------------------------------------------------------
__global__ __launch_bounds__(256) void vq_argmin_kernel(const float* __restrict__ z,
                                                        const float* __restrict__ cb,
                                                        const float* __restrict__ cbnorm,
                                                        int* __restrict__ idx_i,
                                                        float* __restrict__ idx_f) {
  __shared__ float lds_cb[2][TILE_CODES * LDS_STRIDE];

  const int tid  = threadIdx.x;
  const int wave = tid >> 5;
  const int lane = tid & 31;
  const int h    = lane >> 4;   // half-wave selects K pair (A) / K pair (B)
  const int m    = lane & 15;   // A: row M; B/D: column N

  const int row_base = blockIdx.x * ROWS_PER_BLOCK + wave * 16;
  const int row      = row_base + m;              // flat n index (b*1024 + hw)
  const int bb       = row >> 10;
  const int hw       = row & 1023;
  const float* zb    = z + (size_t)bb * (VQ_D * 1024) + hw;

  // A strip in registers: a[j] holds z[M][4j+2h .. 4j+2h+1]  (WMMA 16x4 layout)
  v2f a[64];
#pragma unroll
  for (int j = 0; j < 64; ++j) {
    const int d0 = 4 * j + 2 * h;
    a[j].x = zb[(size_t)d0 * 1024];
    a[j].y = zb[(size_t)(d0 + 1) * 1024];
  }

  float minv[8];
  int   mini[8];
#pragma unroll
  for (int v = 0; v < 8; ++v) { minv[v] = 3.4e38f; mini[v] = 0; }

  // cooperative codebook tile load: thread -> (code lc, 16-float chunk lk)
  const int lc = tid >> 4;
  const int lk = (tid & 15) * 16;
  auto load_tile = [&](int p) {
    const float* src = cb + (size_t)(p * TILE_CODES + lc) * VQ_D + lk;
    float* dst = &lds_cb[p & 1][lc * LDS_STRIDE + lk];
#pragma unroll
    for (int q = 0; q < 16; q += 4) {
      float4 t = *(const float4*)(src + q);
      *(float4*)(dst + q) = t;
    }
  };

  load_tile(0);
  __syncthreads();

  for (int p = 0; p < N_TILES; ++p) {
    if (p + 1 < N_TILES) load_tile(p + 1);                 // double buffer
    if (p + 2 < N_TILES)                                   // global_prefetch_b8
      __builtin_prefetch(cb + (size_t)((p + 2) * TILE_CODES + lc) * VQ_D + lk, 0, 1);

    const float* brow = &lds_cb[p & 1][m * LDS_STRIDE + 2 * h];
    v8f acc = {};
#pragma unroll
    for (int j = 0; j < 64; ++j) {
      v2f bv;                                  // B[k][n] = codebook[n][k]
      bv.x = brow[4 * j];
      bv.y = brow[4 * j + 1];
      acc = __builtin_amdgcn_wmma_f32_16x16x4_f32(
          /*neg_a=*/false, a[j], /*neg_b=*/false, bv,
          /*c_mod=*/(short)0, acc, /*reuse_a=*/false, /*reuse_b=*/false);
    }

    const int   code = p * TILE_CODES + m;     // this lane's column
    const float nrm  = cbnorm[code];
#pragma unroll
    for (int v = 0; v < 8; ++v) {
      const float dst = nrm - 2.0f * acc[v];   // ||e||^2 - 2 z.e
      if (dst < minv[v]) { minv[v] = dst; mini[v] = code; }
    }
    __syncthreads();
  }

  // argmin across the 16 columns held by each 16-lane half
#pragma unroll
  for (int v = 0; v < 8; ++v) {
    float mv = minv[v];
    int   mi = mini[v];
#pragma unroll
    for (int off = 1; off < 16; off <<= 1) {
      const float ov = __shfl_xor(mv, off, 32);
      const int   oi = __shfl_xor(mi, off, 32);
      if (ov < mv || (ov == mv && oi < mi)) { mv = ov; mi = oi; }
    }
    if (m == 0) {
      const int grow = row_base + v + 8 * h;   // D-tile row for (vgpr v, half h)
      idx_i[grow] = mi;
      idx_f[grow] = (float)mi;
    }
  }
}

// ---------------------------------------------------------------------------
// 3) gather z_q (back to B,D,H,W) + fused commit loss
//    grid = B*D = 4096 blocks, 256 threads; block owns one (b,d) slice over hw
// ---------------------------------------------------------------------------
__global__ __launch_bounds__(256) void gather_loss_kernel(const float* __restrict__ z,
                                                          const float* __restrict__ cb,
                                                          const int* __restrict__ idx_i,
                                                          float* __restrict__ zq_out,
                                                          float* __restrict__ loss_out) {
  __shared__ float red[256];
  const int e  = blockIdx.x;        // e = b*256 + d
  const int bb = e >> 8;
  const int d  = e & 255;
  const float* zbd  = z      + (size_t)e * 1024;
  float*       obd  = zq_out + (size_t)e * 1024;
  const int*   idxb = idx_i  + bb * 1024;

  float s = 0.0f;
#pragma unroll
  for (int it = 0; it < 4; ++it) {
    const int hw   = it * 256 + threadIdx.x;
    const int code = idxb[hw];
    const float q  = cb[(size_t)code * VQ_D + d];
    obd[hw] = q;                                   // z_q_st value == z_q
    const float diff = q - zbd[hw];
    s += diff * diff;
  }
  red[threadIdx.x] = s;
  __syncthreads();
#pragma unroll
  for (int o = 128; o > 0; o >>= 1) {
    if (threadIdx.x < o) red[threadIdx.x] += red[threadIdx.x + o];
    __syncthreads();
  }
  if (threadIdx.x == 0)
    atomicAdd(loss_out, red[0] * (0.25f / (float)(VQ_N * VQ_D)));
}

// ---------------------------------------------------------------------------
extern "C" void kernel_launch(void* const* d_in, const int* in_sizes, int n_in,
                              void* d_out, int out_size, void* d_ws, size_t ws_size,
                              hipStream_t stream) {
  (void)in_sizes; (void)n_in; (void)out_size; (void)ws_size;
  const float* z  = (const float*)d_in[0];
  const float* cb = (const float*)d_in[1];

  float* out    = (float*)d_out;
  float* zq     = out;                       // 4,194,304
  float* idx_f  = out + 4194304;             // 16,384
  float* loss   = out + 4194304 + 16384;     // 1

  float* cbnorm = (float*)d_ws;                              // 8192 f32
  int*   idx_i  = (int*)((char*)d_ws + VQ_K * sizeof(float)); // 16384 i32

  hipMemsetAsync(loss, 0, sizeof(float), stream);
  cb_norms_kernel  <<<VQ_K / 8,            256, 0, stream>>>(cb, cbnorm);
  vq_argmin_kernel <<<VQ_N / ROWS_PER_BLOCK, 256, 0, stream>>>(z, cb, cbnorm, idx_i, idx_f);
  gather_loss_kernel<<<16 * VQ_D,          256, 0, stream>>>(z, cb, idx_i, zq, loss);
}